// GraphReasonedLVM_89962384982721
// MI455X (gfx1250) — compile-verified
//
#include <hip/hip_runtime.h>

typedef __attribute__((ext_vector_type(2))) float v2f;
typedef __attribute__((ext_vector_type(8))) float v8f;

#define GAT_ALPHA 0.2f
#define GAT_NEG_INF -9e15f

constexpr int KB = 16;     // K-chunk staged in LDS per pipeline stage
constexpr int BT_LD = 20;  // Bt row stride (floats): 16B-aligned, low-conflict
constexpr int AS_LD = 20;  // As row stride (floats): 16B-aligned, conflict-free

__device__ __forceinline__ v8f wmma_f32(v2f a, v2f b, v8f c) {
  // V_WMMA_F32_16X16X4_F32: D = A(16x4 f32) x B(4x16 f32) + C(16x16 f32)
  return __builtin_amdgcn_wmma_f32_16x16x4_f32(
      /*neg_a=*/false, a, /*neg_b=*/false, b,
      /*c_mod=*/(short)0, c, /*reuse_a=*/false, /*reuse_b=*/false);
}

// ---------------------------------------------------------------------------
// x[t][n][c] = sum_d pose[t][n][d] * Wp[d][c] + bp[c]     (Din = 3)
// ---------------------------------------------------------------------------
__global__ void proj_kernel(const float* __restrict__ pose,
                            const float* __restrict__ Wp,
                            const float* __restrict__ bp,
                            float* __restrict__ x) {
  int t = blockIdx.x, n = blockIdx.y, c = threadIdx.x;
  const float* p = pose + ((long)t * 512 + n) * 3;
  float acc = bp[c];
  acc += p[0] * Wp[0 * 256 + c];
  acc += p[1] * Wp[1 * 256 + c];
  acc += p[2] * Wp[2 * 256 + c];
  x[((long)t * 512 + n) * 256 + c] = acc;
}

// ---------------------------------------------------------------------------
// fp32 WMMA GEMM: C[512 x 256] = A[512 x K] @ B[K x 256]  (ldb = ldc = 256)
// grid = (T, nHeads, 512/16); block = 256 (8 waves); wave -> 16 x 32 tile.
// B transposed into LDS (Bt[col][k]) so each WMMA B operand is one
// ds_load_b64 into a contiguous register pair. Register-prefetch pipeline.
// ---------------------------------------------------------------------------
__global__ __launch_bounds__(256) void wmma_gemm_kernel(
    const float* __restrict__ A, long a_frame_stride, int lda,
    const float* __restrict__ B, long b_head_stride,
    float* __restrict__ C, long c_frame_stride, long c_head_stride, int K) {
  __shared__ float As[16][AS_LD];   // A chunk: 16 rows x KB
  __shared__ float Bt[256][BT_LD];  // B chunk transposed: Bt[col][k]

  int t = blockIdx.x, h = blockIdx.y, mt = blockIdx.z;
  int m0 = mt * 16;
  const float* Ab = A + (long)t * a_frame_stride + (long)m0 * lda;
  const float* Bb = B + (long)h * b_head_stride;
  float* Cb = C + (long)t * c_frame_stride + (long)h * c_head_stride;

  int tid = threadIdx.x;
  int lane = tid & 31, wave = tid >> 5;
  int n0 = wave * 32;
  int mrow = lane & 15;       // A row / B-C column inside the tile
  int kh = (lane >> 4) << 1;  // K sub-offset per lane half: 0 or 2

  // B staging: one column per thread; 16 coalesced b32 loads per chunk.
  const float* bCol = Bb + tid;
  // A staging: threads 0..63 take one float4 each (16 x KB chunk).
  int arow = tid >> 2, akq = tid & 3;
  const float* aG = Ab + (long)arow * lda + akq * 4;

  float4 bReg[4];
  float4 aReg = {0.f, 0.f, 0.f, 0.f};
  int nChunks = K / KB;

  // prologue: prefetch chunk 0
#pragma unroll
  for (int kq = 0; kq < 4; ++kq) {
    bReg[kq].x = bCol[(long)(kq * 4 + 0) * 256];
    bReg[kq].y = bCol[(long)(kq * 4 + 1) * 256];
    bReg[kq].z = bCol[(long)(kq * 4 + 2) * 256];
    bReg[kq].w = bCol[(long)(kq * 4 + 3) * 256];
  }
  if (tid < 64) aReg = *(const float4*)aG;

  v8f acc[2] = {};
  for (int c = 0; c < nChunks; ++c) {
    __syncthreads();  // previous chunk fully consumed
#pragma unroll
    for (int kq = 0; kq < 4; ++kq) *(float4*)&Bt[tid][kq * 4] = bReg[kq];
    if (tid < 64) *(float4*)&As[arow][akq * 4] = aReg;
    // prefetch next chunk: overlaps the WMMA burst below
    if (c + 1 < nChunks) {
      const float* bNext = bCol + (long)(c + 1) * (KB * 256);
#pragma unroll
      for (int kq = 0; kq < 4; ++kq) {
        bReg[kq].x = bNext[(long)(kq * 4 + 0) * 256];
        bReg[kq].y = bNext[(long)(kq * 4 + 1) * 256];
        bReg[kq].z = bNext[(long)(kq * 4 + 2) * 256];
        bReg[kq].w = bNext[(long)(kq * 4 + 3) * 256];
      }
      if (tid < 64) aReg = *(const float4*)(aG + (c + 1) * KB);
    }
    __syncthreads();  // chunk staged
#pragma unroll
    for (int k0 = 0; k0 < KB; k0 += 4) {
      v2f a = *(const v2f*)&As[mrow][k0 + kh];
#pragma unroll
      for (int nt = 0; nt < 2; ++nt) {
        v2f b = *(const v2f*)&Bt[n0 + nt * 16 + mrow][k0 + kh];
        acc[nt] = wmma_f32(a, b, acc[nt]);
      }
    }
  }

#pragma unroll
  for (int nt = 0; nt < 2; ++nt)
#pragma unroll
    for (int r = 0; r < 8; ++r) {
      int row = m0 + r + ((lane >> 4) << 3);  // M = r (+8 for upper half)
      Cb[(long)row * 256 + n0 + nt * 16 + mrow] = acc[nt][r];
    }
}

// ---------------------------------------------------------------------------
// f[i] = Wh[i,:] . a1 ; g[i] = Wh[i,:] . a2   (a1 = a[:256], a2 = a[256:])
// grid = (T, nHeads, 64); block = 256 (8 waves); wave-per-row, coalesced.
// ---------------------------------------------------------------------------
__global__ void fg_kernel(const float* __restrict__ Wh, long wh_frame_stride,
                          long wh_head_stride, const float* __restrict__ a_vec,
                          long a_head_stride, float* __restrict__ f,
                          float* __restrict__ g, long fg_frame_stride,
                          long fg_head_stride) {
  int t = blockIdx.x, h = blockIdx.y;
  int wave = threadIdx.x >> 5, lane = threadIdx.x & 31;
  int row = blockIdx.z * 8 + wave;
  const float* w = Wh + (long)t * wh_frame_stride + (long)h * wh_head_stride +
                   (long)row * 256;
  const float* a1 = a_vec + (long)h * a_head_stride;
  const float* a2 = a1 + 256;
  float sf = 0.f, sg = 0.f;
#pragma unroll
  for (int c0 = 0; c0 < 256; c0 += 32) {
    float v = w[c0 + lane];
    sf += v * a1[c0 + lane];
    sg += v * a2[c0 + lane];
  }
#pragma unroll
  for (int m = 16; m; m >>= 1) {
    sf += __shfl_xor(sf, m, 32);
    sg += __shfl_xor(sg, m, 32);
  }
  if (lane == 0) {
    long o = (long)t * fg_frame_stride + (long)h * fg_head_stride + row;
    f[o] = sf;
    g[o] = sg;
  }
}

// ---------------------------------------------------------------------------
// Fused attention: 16-row softmax tile in LDS, then LDS-staged WMMA GEMM
// attn[16x512] @ Wh[512x256], with 1/rowsum + ELU folded into the epilogue.
// grid = (T, nHeads, 512/16); block = 256 (8 waves).
// ---------------------------------------------------------------------------
__global__ __launch_bounds__(256) void attn_kernel(
    const float* __restrict__ Wh, long wh_frame_stride, long wh_head_stride,
    const float* __restrict__ f, const float* __restrict__ g,
    long fg_frame_stride, long fg_head_stride, const int* __restrict__ adj,
    float* __restrict__ out, long out_frame_stride, int out_ld,
    int out_col_head_stride, int double_elu) {
  __shared__ float p[16][516];      // softmax tile (A operand)
  __shared__ float Bt[256][BT_LD];  // Wh chunk transposed: Bt[col][k]
  __shared__ float gs[512];
  __shared__ float red[16][16];
  __shared__ float rowmax[16];
  __shared__ float rowinv[16];

  int t = blockIdx.x, h = blockIdx.y, mt = blockIdx.z;
  int m0 = mt * 16;
  const float* Whb = Wh + (long)t * wh_frame_stride + (long)h * wh_head_stride;
  const float* fb = f + (long)t * fg_frame_stride + (long)h * fg_head_stride;
  const float* gb = g + (long)t * fg_frame_stride + (long)h * fg_head_stride;
  int tid = threadIdx.x;

  for (int j = tid; j < 512; j += 256) gs[j] = gb[j];
  __syncthreads();

  int r = tid >> 4;       // row 0..15
  int cchunk = tid & 15;  // 32-column chunk
  float fi = fb[m0 + r];
  const int* adjrow = adj + (long)(m0 + r) * 512;

  // pass 1: masked leaky-relu logits + row max
  float mx = -3.0e38f;
  for (int j = cchunk * 32; j < cchunk * 32 + 32; ++j) {
    float e = fi + gs[j];
    e = e > 0.f ? e : GAT_ALPHA * e;
    e = adjrow[j] > 0 ? e : GAT_NEG_INF;
    p[r][j] = e;
    mx = fmaxf(mx, e);
  }
  red[r][cchunk] = mx;
  __syncthreads();
  if (tid < 16) {
    float m2 = red[tid][0];
    for (int c = 1; c < 16; ++c) m2 = fmaxf(m2, red[tid][c]);
    rowmax[tid] = m2;
  }
  __syncthreads();

  // pass 2: exp + row sum (normalization deferred to epilogue)
  float m2 = rowmax[r];
  float s = 0.f;
  for (int j = cchunk * 32; j < cchunk * 32 + 32; ++j) {
    float pe = __expf(p[r][j] - m2);
    p[r][j] = pe;
    s += pe;
  }
  red[r][cchunk] = s;
  __syncthreads();
  if (tid < 16) {
    float s2 = 0.f;
    for (int c = 0; c < 16; ++c) s2 += red[tid][c];
    rowinv[tid] = 1.0f / s2;
  }
  // (ordering of p/rowinv vs consumers handled by the loop's first barrier)

  // WMMA GEMM: [16 x 512] (LDS) @ Wh [512 x 256], B LDS-staged + prefetched
  int lane = tid & 31, wave = tid >> 5;
  int n0 = wave * 32;
  int mrow = lane & 15;
  int kh = (lane >> 4) << 1;

  const float* bCol = Whb + tid;
  float4 bReg[4];
#pragma unroll
  for (int kq = 0; kq < 4; ++kq) {
    bReg[kq].x = bCol[(long)(kq * 4 + 0) * 256];
    bReg[kq].y = bCol[(long)(kq * 4 + 1) * 256];
    bReg[kq].z = bCol[(long)(kq * 4 + 2) * 256];
    bReg[kq].w = bCol[(long)(kq * 4 + 3) * 256];
  }

  v8f acc[2] = {};
  const int nChunks = 512 / KB;  // 32
  for (int c = 0; c < nChunks; ++c) {
    __syncthreads();
#pragma unroll
    for (int kq = 0; kq < 4; ++kq) *(float4*)&Bt[tid][kq * 4] = bReg[kq];
    if (c + 1 < nChunks) {
      const float* bNext = bCol + (long)(c + 1) * (KB * 256);
#pragma unroll
      for (int kq = 0; kq < 4; ++kq) {
        bReg[kq].x = bNext[(long)(kq * 4 + 0) * 256];
        bReg[kq].y = bNext[(long)(kq * 4 + 1) * 256];
        bReg[kq].z = bNext[(long)(kq * 4 + 2) * 256];
        bReg[kq].w = bNext[(long)(kq * 4 + 3) * 256];
      }
    }
    __syncthreads();
#pragma unroll
    for (int k0 = 0; k0 < KB; k0 += 4) {
      v2f a = *(const v2f*)&p[mrow][c * KB + k0 + kh];
#pragma unroll
      for (int nt = 0; nt < 2; ++nt) {
        v2f b = *(const v2f*)&Bt[n0 + nt * 16 + mrow][k0 + kh];
        acc[nt] = wmma_f32(a, b, acc[nt]);
      }
    }
  }

  float* ob = out + (long)t * out_frame_stride + (long)h * out_col_head_stride;
#pragma unroll
  for (int nt = 0; nt < 2; ++nt)
#pragma unroll
    for (int r2 = 0; r2 < 8; ++r2) {
      int row = r2 + ((lane >> 4) << 3);
      float v = acc[nt][r2] * rowinv[row];
      v = v > 0.f ? v : __expf(v) - 1.f;                  // elu
      if (double_elu) v = v > 0.f ? v : __expf(v) - 1.f;  // elu(elu(.))
      ob[(long)(m0 + row) * out_ld + n0 + nt * 16 + mrow] = v;
    }
}

// ---------------------------------------------------------------------------
// out[t] = mean_n(x[t]) @ Wo + bo
// ---------------------------------------------------------------------------
__global__ void pool_kernel(const float* __restrict__ x,
                            const float* __restrict__ Wo,
                            const float* __restrict__ bo,
                            float* __restrict__ out) {
  __shared__ float pooled[256];
  int t = blockIdx.x, c = threadIdx.x;
  const float* xb = x + (long)t * 512 * 256;
  float s = 0.f;
  for (int n = 0; n < 512; ++n) s += xb[(long)n * 256 + c];
  pooled[c] = s * (1.0f / 512.0f);
  __syncthreads();
  float acc = bo[c];
  for (int k = 0; k < 256; ++k) acc += pooled[k] * Wo[k * 256 + c];
  out[(long)t * 256 + c] = acc;
}

// ---------------------------------------------------------------------------
extern "C" void kernel_launch(void* const* d_in, const int* in_sizes, int n_in,
                              void* d_out, int out_size, void* d_ws,
                              size_t ws_size, hipStream_t stream) {
  const float* pose = (const float*)d_in[0];     // [32,512,3]
  const int* adj = (const int*)d_in[1];          // [512,512]
  const float* Wp = (const float*)d_in[2];       // [3,256]
  const float* bp = (const float*)d_in[3];       // [256]
  const float* W_heads = (const float*)d_in[4];  // [3,8,256,256]
  const float* a_heads = (const float*)d_in[5];  // [3,8,512]
  const float* W_out = (const float*)d_in[6];    // [3,2048,256]
  const float* a_out = (const float*)d_in[7];    // [3,512]
  const float* Wo = (const float*)d_in[8];       // [256,256]
  const float* bo = (const float*)d_in[9];       // [256]
  float* out = (float*)d_out;                    // [32,256]

  const int T = 32, N = 512, F = 256, H = 8, L = 3;
  char* ws = (char*)d_ws;
  size_t off = 0;
  float* x = (float*)(ws + off);     off += (size_t)T * N * F * 4;
  float* xcat = (float*)(ws + off);  off += (size_t)T * N * H * F * 4;
  float* Wh_h = (float*)(ws + off);  off += (size_t)T * H * N * F * 4;
  float* Wh_o = (float*)(ws + off);  off += (size_t)T * N * F * 4;
  float* f_h = (float*)(ws + off);   off += (size_t)T * H * N * 4;
  float* g_h = (float*)(ws + off);   off += (size_t)T * H * N * 4;
  float* f_o = (float*)(ws + off);   off += (size_t)T * N * 4;
  float* g_o = (float*)(ws + off);   off += (size_t)T * N * 4;

  proj_kernel<<<dim3(T, N), 256, 0, stream>>>(pose, Wp, bp, x);

  for (int l = 0; l < L; ++l) {
    // per-head projections: Wh_h[t][h] = x[t] @ W_heads[l][h]
    wmma_gemm_kernel<<<dim3(T, H, N / 16), 256, 0, stream>>>(
        x, (long)N * F, F,
        W_heads + (size_t)l * H * F * F, (long)F * F,
        Wh_h, (long)H * N * F, (long)N * F, F);
    fg_kernel<<<dim3(T, H, 64), 256, 0, stream>>>(
        Wh_h, (long)H * N * F, (long)N * F,
        a_heads + (size_t)l * H * 2 * F, (long)2 * F,
        f_h, g_h, (long)H * N, (long)N);
    // per-head attention -> concat into xcat columns [h*256, h*256+256)
    attn_kernel<<<dim3(T, H, N / 16), 256, 0, stream>>>(
        Wh_h, (long)H * N * F, (long)N * F,
        f_h, g_h, (long)H * N, (long)N, adj,
        xcat, (long)N * H * F, H * F, F, /*double_elu=*/0);
    // out attention: Wh_o[t] = xcat[t] @ W_out[l]   (K = 2048)
    wmma_gemm_kernel<<<dim3(T, 1, N / 16), 256, 0, stream>>>(
        xcat, (long)N * H * F, H * F,
        W_out + (size_t)l * H * F * F, 0,
        Wh_o, (long)N * F, 0, H * F);
    fg_kernel<<<dim3(T, 1, 64), 256, 0, stream>>>(
        Wh_o, (long)N * F, 0,
        a_out + (size_t)l * 2 * F, 0,
        f_o, g_o, (long)N, 0);
    attn_kernel<<<dim3(T, 1, N / 16), 256, 0, stream>>>(
        Wh_o, (long)N * F, 0,
        f_o, g_o, (long)N, 0, adj,
        x, (long)N * F, F, 0, /*double_elu=*/1);
  }

  pool_kernel<<<T, 256, 0, stream>>>(x, Wo, bo, out);
}